// MaskedSoftmax_21492016349220
// MI455X (gfx1250) — compile-verified
//
#include <hip/hip_runtime.h>
#include <stdint.h>

// Masked row softmax: 4096 rows x 8192 cols, fp32 input, int32 mask, fp32 out.
// Memory-bound (384 MB @ 23.3 TB/s ~= 16.5us floor). Strategy: one workgroup
// per row; stage row (x + mask) into LDS with CDNA5 async global->LDS DMA
// (non-temporal: zero reuse, keep L2 clean), two register-resident passes
// (max, exp+sum), coalesced non-temporal b128 stores.

#define ROW_LEN  8192
#define THREADS  256
#define V4PT     (ROW_LEN / (THREADS * 4))   // 8 float4 per thread
#define NWAVES   (THREADS / 32)              // 8 wave32 per workgroup

typedef float v4f __attribute__((ext_vector_type(4)));

// CDNA5 async copy: global memory -> LDS, 16 bytes per lane, tracked by
// ASYNCcnt. TH_LOAD_NT: streamed-once data, don't pollute caches.
__device__ __forceinline__ void async_load_b128_nt(uint32_t lds_byte_off, const void* gptr) {
    asm volatile("global_load_async_to_lds_b128 %0, %1, off th:TH_LOAD_NT"
                 :
                 : "v"(lds_byte_off), "v"((uint64_t)(uintptr_t)gptr)
                 : "memory");
}

__device__ __forceinline__ void wait_async_all() {
    asm volatile("s_wait_asynccnt 0x0" ::: "memory");
}

__device__ __forceinline__ float wave_reduce_max(float v) {
#pragma unroll
    for (int off = 16; off > 0; off >>= 1)
        v = fmaxf(v, __shfl_xor(v, off, 32));
    return v;
}

__device__ __forceinline__ float wave_reduce_sum(float v) {
#pragma unroll
    for (int off = 16; off > 0; off >>= 1)
        v += __shfl_xor(v, off, 32);
    return v;
}

__global__ __launch_bounds__(THREADS)
void MaskedSoftmax_rows_kernel(const float* __restrict__ x,
                               const int*   __restrict__ mask,
                               float*       __restrict__ out)
{
    __shared__ __align__(16) float sX[ROW_LEN];   // 32 KB
    __shared__ __align__(16) int   sM[ROW_LEN];   // 32 KB
    __shared__ float sRed[2 * NWAVES];

    const int    tid  = threadIdx.x;
    const size_t base = (size_t)blockIdx.x * ROW_LEN;

    // LDS byte offsets of the staging buffers (flat addr low 32 bits == LDS offset).
    const uint32_t sx0 = (uint32_t)(uintptr_t)(void*)sX;
    const uint32_t sm0 = (uint32_t)(uintptr_t)(void*)sM;

    // ---- Stage the whole row into LDS via async DMA (ASYNCcnt) -------------
#pragma unroll
    for (int i = 0; i < V4PT; ++i) {
        const uint32_t e = (uint32_t)(i * THREADS + tid) * 4u;  // element index
        async_load_b128_nt(sx0 + e * 4u, x    + base + e);
        async_load_b128_nt(sm0 + e * 4u, mask + base + e);
    }
    wait_async_all();      // this wave's DMA done (data resident in LDS)
    __syncthreads();       // all waves' DMA done

    const float4* sX4 = (const float4*)sX;
    const int4*   sM4 = (const int4*)sM;

    // ---- Pass 1: masked values -> registers, local max ---------------------
    float4 w[V4PT];
    float  lmax = -INFINITY;
#pragma unroll
    for (int i = 0; i < V4PT; ++i) {
        const int    v  = i * THREADS + tid;
        const float4 xv = sX4[v];
        const int4   mv = sM4[v];
        float4 t;
        t.x = (mv.x == 1) ? xv.x : -INFINITY;
        t.y = (mv.y == 1) ? xv.y : -INFINITY;
        t.z = (mv.z == 1) ? xv.z : -INFINITY;
        t.w = (mv.w == 1) ? xv.w : -INFINITY;
        w[i] = t;
        lmax = fmaxf(lmax, fmaxf(fmaxf(t.x, t.y), fmaxf(t.z, t.w)));
    }

    lmax = wave_reduce_max(lmax);
    const int wave = tid >> 5;
    if ((tid & 31) == 0) sRed[wave] = lmax;
    __syncthreads();
    float mx = sRed[0];
#pragma unroll
    for (int wv = 1; wv < NWAVES; ++wv) mx = fmaxf(mx, sRed[wv]);
    // Fully-masked row guard: max over empty set -> use 0 (matches reference).
    const float mxs = (mx == -INFINITY) ? 0.0f : mx;

    // ---- Pass 2: e = exp(xm - mxs) (exp(-inf)=0 handles masked), local sum --
    float lsum = 0.0f;
#pragma unroll
    for (int i = 0; i < V4PT; ++i) {
        float4 t = w[i];
        float4 ev;
        ev.x = expf(t.x - mxs);
        ev.y = expf(t.y - mxs);
        ev.z = expf(t.z - mxs);
        ev.w = expf(t.w - mxs);
        w[i] = ev;
        lsum += (ev.x + ev.y) + (ev.z + ev.w);
    }

    lsum = wave_reduce_sum(lsum);
    if ((tid & 31) == 0) sRed[NWAVES + wave] = lsum;
    __syncthreads();
    float s = 0.0f;
#pragma unroll
    for (int wv = 0; wv < NWAVES; ++wv) s += sRed[NWAVES + wv];
    const float r = (s == 0.0f) ? 0.0f : 1.0f / s;   // s==0 => all e==0 => zeros

    // ---- Pass 3: normalize + coalesced non-temporal b128 stores -------------
    v4f* o4 = (v4f*)(out + base);
#pragma unroll
    for (int i = 0; i < V4PT; ++i) {
        const float4 ev = w[i];
        v4f rv;
        rv.x = ev.x * r;
        rv.y = ev.y * r;
        rv.z = ev.z * r;
        rv.w = ev.w * r;
        __builtin_nontemporal_store(rv, &o4[i * THREADS + tid]);
    }
}

extern "C" void kernel_launch(void* const* d_in, const int* in_sizes, int n_in,
                              void* d_out, int out_size, void* d_ws, size_t ws_size,
                              hipStream_t stream) {
    (void)n_in; (void)out_size; (void)d_ws; (void)ws_size;
    const float* x    = (const float*)d_in[0];
    const int*   mask = (const int*)d_in[1];
    float*       out  = (float*)d_out;

    const int rows = in_sizes[0] / ROW_LEN;   // 4096
    MaskedSoftmax_rows_kernel<<<rows, THREADS, 0, stream>>>(x, mask, out);
}